// DotProductAttention_73452530696353
// MI455X (gfx1250) — compile-verified
//
#include <hip/hip_runtime.h>

typedef float v2f __attribute__((ext_vector_type(2)));
typedef float v8f __attribute__((ext_vector_type(8)));

#define BATCH 16
#define LSEQ  2048
#define DDIM  128
#define TILE  128
#define NT    (LSEQ / TILE)   // 16 tiles per dimension
#define EPS   1e-12f
#define NEG_INF (-__builtin_inff())

// Order-independent (exact) float atomic max via signed/unsigned int trick.
__device__ __forceinline__ void atomicMaxF(float* addr, float val) {
    if (val >= 0.0f) atomicMax((int*)addr, __float_as_int(val));
    else             atomicMin((unsigned int*)addr, __float_as_uint(val));
}

__global__ __launch_bounds__(256) void k_init_stats(float* rowmax, float* colmax) {
    int i = blockIdx.x * blockDim.x + threadIdx.x;
    if (i < BATCH * LSEQ) {
        rowmax[i] = NEG_INF;
        colmax[i] = NEG_INF;
    }
}

// ---------------------------------------------------------------------------
// Pass 1: S = q @ k^T per 128x128 tile via v_wmma_f32_16x16x4_f32.
// Writes raw scores S into the gated output region; computes row/col maxima.
// ---------------------------------------------------------------------------
__global__ __launch_bounds__(256) void k_qk(const float* __restrict__ q,
                                            const float* __restrict__ k,
                                            float* __restrict__ sbuf,
                                            float* __restrict__ rowmax,
                                            float* __restrict__ colmax) {
    __shared__ float ksh[TILE][65];   // k-tile half (128 x 64), +1 pad vs bank conflicts
    __shared__ float colred[TILE];

    const int bid = blockIdx.x;
    const int b   = bid / (NT * NT);
    const int it  = (bid / NT) % NT;
    const int jt  = bid % NT;
    const int i0  = it * TILE, j0 = jt * TILE;

    const int tid  = threadIdx.x;
    const int wave = tid >> 5;        // 8 waves, each owns a 16-row strip
    const int lane = tid & 31;
    const int m    = lane & 15;
    const int hi   = lane >> 4;

    const float* qb = q + (size_t)b * LSEQ * DDIM;
    const float* kb = k + (size_t)b * LSEQ * DDIM;

    if (tid < TILE) colred[tid] = NEG_INF;

    v8f zero = {};
    v8f acc[8];
    #pragma unroll
    for (int js = 0; js < 8; ++js) acc[js] = zero;

    const int arow = i0 + wave * 16 + m;   // A-fragment row for this lane

    for (int half = 0; half < 2; ++half) {
        __syncthreads();
        // Stage k tile half into LDS: rows j0..j0+127, cols half*64..+63
        for (int idx = tid; idx < TILE * 16; idx += 256) {
            const int r  = idx >> 4;
            const int c4 = (idx & 15) << 2;
            const float4 val =
                *(const float4*)(kb + (size_t)(j0 + r) * DDIM + half * 64 + c4);
            ksh[r][c4 + 0] = val.x;
            ksh[r][c4 + 1] = val.y;
            ksh[r][c4 + 2] = val.z;
            ksh[r][c4 + 3] = val.w;
        }
        __syncthreads();

        for (int s = 0; s < 16; ++s) {
            const int kd = s * 4 + hi * 2;   // K offset within this half
            const float2 av =
                *(const float2*)(qb + (size_t)arow * DDIM + half * 64 + kd);
            v2f a; a.x = av.x; a.y = av.y;
            // Load ALL B-fragments first so the ds_loads batch & pipeline,
            // then issue the 8 WMMAs back-to-back behind a single wait.
            v2f bm[8];
            #pragma unroll
            for (int js = 0; js < 8; ++js) {
                bm[js].x = ksh[js * 16 + m][kd];
                bm[js].y = ksh[js * 16 + m][kd + 1];
            }
            #pragma unroll
            for (int js = 0; js < 8; ++js)
                acc[js] = __builtin_amdgcn_wmma_f32_16x16x4_f32(
                    false, a, false, bm[js], (short)0, acc[js], false, false);
        }
    }

    // Store S tile + per-row maxima (wave fully owns its 16 rows).
    #pragma unroll
    for (int r = 0; r < 8; ++r) {
        const int row = i0 + wave * 16 + r + hi * 8;
        float* sp = sbuf + ((size_t)b * LSEQ + row) * LSEQ + j0;
        float rmax = NEG_INF;
        #pragma unroll
        for (int js = 0; js < 8; ++js) {
            const float vv = acc[js][r];
            sp[js * 16 + m] = vv;
            rmax = fmaxf(rmax, vv);
        }
        #pragma unroll
        for (int off = 1; off < 16; off <<= 1)
            rmax = fmaxf(rmax, __shfl_xor(rmax, off, 32));
        if (m == 0) atomicMaxF(&rowmax[b * LSEQ + row], rmax);
    }

    // Per-column maxima: reduce across regs, lane halves, then waves via LDS.
    #pragma unroll
    for (int js = 0; js < 8; ++js) {
        float cmax = NEG_INF;
        #pragma unroll
        for (int r = 0; r < 8; ++r) cmax = fmaxf(cmax, acc[js][r]);
        cmax = fmaxf(cmax, __shfl_xor(cmax, 16, 32));
        if (hi == 0) atomicMaxF(&colred[js * 16 + m], cmax);
    }
    __syncthreads();
    if (tid < TILE) atomicMaxF(&colmax[b * LSEQ + j0 + tid], colred[tid]);
}

// ---------------------------------------------------------------------------
// Pass 2: per-tile partial sums of exp(S - max) * mask for rows and columns.
// Deterministic: partials land in private (b,row,jtile)/(b,col,itile) slots.
// ---------------------------------------------------------------------------
__global__ __launch_bounds__(256) void k_sums(const float* __restrict__ sbuf,
                                              const float* __restrict__ mask,
                                              const float* __restrict__ rowmax,
                                              const float* __restrict__ colmax,
                                              float* __restrict__ rowpart,
                                              float* __restrict__ colpart) {
    __shared__ float csh[8][TILE];
    const int bid = blockIdx.x;
    const int b   = bid / (NT * NT);
    const int it  = (bid / NT) % NT;
    const int jt  = bid % NT;
    const int i0  = it * TILE, j0 = jt * TILE;

    const int tid  = threadIdx.x;
    const int wave = tid >> 5;
    const int lane = tid & 31;

    float cm[4], ca[4];
    #pragma unroll
    for (int ch = 0; ch < 4; ++ch) {
        cm[ch] = colmax[b * LSEQ + j0 + ch * 32 + lane];
        ca[ch] = 0.0f;
    }

    for (int r = 0; r < 16; ++r) {
        const int row = i0 + wave * 16 + r;
        const float rm = rowmax[b * LSEQ + row];
        const float* sp = sbuf + ((size_t)b * LSEQ + row) * LSEQ + j0;
        const float* mp = mask + ((size_t)b * LSEQ + row) * LSEQ + j0;
        float rpart = 0.0f;
        #pragma unroll
        for (int ch = 0; ch < 4; ++ch) {
            const int c = ch * 32 + lane;
            const float sv = sp[c];
            const float mv = mp[c];
            rpart  += __expf(sv - rm) * mv;
            ca[ch] += __expf(sv - cm[ch]) * mv;
        }
        #pragma unroll
        for (int off = 1; off < 32; off <<= 1)
            rpart += __shfl_xor(rpart, off, 32);
        if (lane == 0)
            rowpart[((size_t)b * LSEQ + row) * NT + jt] = rpart;
    }

    #pragma unroll
    for (int ch = 0; ch < 4; ++ch) csh[wave][ch * 32 + lane] = ca[ch];
    __syncthreads();
    if (tid < TILE) {
        float s = 0.0f;
        #pragma unroll
        for (int w = 0; w < 8; ++w) s += csh[w][tid];
        colpart[((size_t)b * LSEQ + j0 + tid) * NT + it] = s;
    }
}

__global__ __launch_bounds__(256) void k_sums2(const float* __restrict__ rowpart,
                                               const float* __restrict__ colpart,
                                               float* __restrict__ rowsum,
                                               float* __restrict__ colsum) {
    int i = blockIdx.x * blockDim.x + threadIdx.x;
    if (i < BATCH * LSEQ) {
        float rs = 0.0f, cs = 0.0f;
        #pragma unroll
        for (int t = 0; t < NT; ++t) {
            rs += rowpart[(size_t)i * NT + t];
            cs += colpart[(size_t)i * NT + t];
        }
        rowsum[i] = rs;
        colsum[i] = cs;
    }
}

// ---------------------------------------------------------------------------
// Pass 3: gated = m * exp(2S - rm - cm) / ((rs+eps)(cs+eps)) written in place,
// and O = gated @ v accumulated with v_wmma_f32_16x16x4_f32 (A-frags via LDS).
// Each workgroup owns a full 128-row strip -> no atomics on the output.
// ---------------------------------------------------------------------------
__global__ __launch_bounds__(256) void k_final(const float* __restrict__ v,
                                               const float* __restrict__ mask,
                                               float* __restrict__ gated,   // holds S on entry
                                               const float* __restrict__ rowmax,
                                               const float* __restrict__ colmax,
                                               const float* __restrict__ rowsum,
                                               const float* __restrict__ colsum,
                                               float* __restrict__ out) {
    __shared__ float gsh[TILE][65];   // gated tile half (128 x 64)
    const int bid = blockIdx.x;
    const int b   = bid / NT;
    const int it  = bid % NT;
    const int i0  = it * TILE;

    const int tid  = threadIdx.x;
    const int wave = tid >> 5;
    const int lane = tid & 31;
    const int m    = lane & 15;
    const int hi   = lane >> 4;

    const float* vb = v     + (size_t)b * LSEQ * DDIM;
    float*       gb = gated + (size_t)b * LSEQ * LSEQ;
    const float* mb = mask  + (size_t)b * LSEQ * LSEQ;

    v8f zero = {};
    v8f acc[8];
    #pragma unroll
    for (int d8 = 0; d8 < 8; ++d8) acc[d8] = zero;

    for (int jt = 0; jt < NT; ++jt) {
        for (int half = 0; half < 2; ++half) {
            const int c0 = jt * TILE + half * 64;
            __syncthreads();
            // Coalesced: read S + mask, compute gated, write back, stash in LDS.
            for (int idx = tid; idx < TILE * 16; idx += 256) {
                const int r = idx >> 4;
                const int c = (idx & 15) << 2;
                const int grow = i0 + r;
                float* gp = gb + (size_t)grow * LSEQ + c0 + c;
                const float* mp = mb + (size_t)grow * LSEQ + c0 + c;
                const float4 s4 = *(const float4*)gp;
                const float4 m4 = *(const float4*)mp;
                const float rm = rowmax[b * LSEQ + grow];
                const float ri = 1.0f / (rowsum[b * LSEQ + grow] + EPS);
                float4 g4;
                g4.x = m4.x * __expf(2.0f * s4.x - rm - colmax[b * LSEQ + c0 + c + 0]) * ri
                       / (colsum[b * LSEQ + c0 + c + 0] + EPS);
                g4.y = m4.y * __expf(2.0f * s4.y - rm - colmax[b * LSEQ + c0 + c + 1]) * ri
                       / (colsum[b * LSEQ + c0 + c + 1] + EPS);
                g4.z = m4.z * __expf(2.0f * s4.z - rm - colmax[b * LSEQ + c0 + c + 2]) * ri
                       / (colsum[b * LSEQ + c0 + c + 2] + EPS);
                g4.w = m4.w * __expf(2.0f * s4.w - rm - colmax[b * LSEQ + c0 + c + 3]) * ri
                       / (colsum[b * LSEQ + c0 + c + 3] + EPS);
                *(float4*)gp = g4;
                gsh[r][c + 0] = g4.x;
                gsh[r][c + 1] = g4.y;
                gsh[r][c + 2] = g4.z;
                gsh[r][c + 3] = g4.w;
            }
            __syncthreads();

            // O += gated_half (128 x 64) @ v_half (64 x 128)
            for (int s = 0; s < 16; ++s) {
                const int kd = s * 4 + hi * 2;
                v2f a;
                a.x = gsh[wave * 16 + m][kd];
                a.y = gsh[wave * 16 + m][kd + 1];
                // Batch all v-fragment loads ahead of the WMMA chain.
                v2f bm[8];
                #pragma unroll
                for (int d8 = 0; d8 < 8; ++d8) {
                    bm[d8].x = vb[(size_t)(c0 + kd)     * DDIM + d8 * 16 + m];
                    bm[d8].y = vb[(size_t)(c0 + kd + 1) * DDIM + d8 * 16 + m];
                }
                #pragma unroll
                for (int d8 = 0; d8 < 8; ++d8)
                    acc[d8] = __builtin_amdgcn_wmma_f32_16x16x4_f32(
                        false, a, false, bm[d8], (short)0, acc[d8], false, false);
            }
        }
    }

    #pragma unroll
    for (int r = 0; r < 8; ++r) {
        const int orow = i0 + wave * 16 + r + hi * 8;
        #pragma unroll
        for (int d8 = 0; d8 < 8; ++d8)
            out[((size_t)b * LSEQ + orow) * DDIM + d8 * 16 + m] = acc[d8][r];
    }
}

extern "C" void kernel_launch(void* const* d_in, const int* in_sizes, int n_in,
                              void* d_out, int out_size, void* d_ws, size_t ws_size,
                              hipStream_t stream) {
    (void)in_sizes; (void)n_in; (void)out_size; (void)ws_size;
    const float* q    = (const float*)d_in[0];
    const float* k    = (const float*)d_in[1];
    const float* v    = (const float*)d_in[2];
    const float* mask = (const float*)d_in[3];

    float* out   = (float*)d_out;                          // [B, Lq, D]
    float* gated = out + (size_t)BATCH * LSEQ * DDIM;      // [B, Lq, Lk]

    const size_t BL = (size_t)BATCH * LSEQ;
    float* rowmax  = (float*)d_ws;          // BL
    float* colmax  = rowmax + BL;           // BL
    float* rowsum  = colmax + BL;           // BL
    float* colsum  = rowsum + BL;           // BL
    float* rowpart = colsum + BL;           // BL * NT
    float* colpart = rowpart + BL * NT;     // BL * NT   (total ~4.7 MB)

    k_init_stats<<<dim3((BATCH * LSEQ + 255) / 256), dim3(256), 0, stream>>>(rowmax, colmax);
    k_qk<<<dim3(BATCH * NT * NT), dim3(256), 0, stream>>>(q, k, gated, rowmax, colmax);
    k_sums<<<dim3(BATCH * NT * NT), dim3(256), 0, stream>>>(gated, mask, rowmax, colmax,
                                                            rowpart, colpart);
    k_sums2<<<dim3((BATCH * LSEQ + 255) / 256), dim3(256), 0, stream>>>(rowpart, colpart,
                                                                        rowsum, colsum);
    k_final<<<dim3(BATCH * NT), dim3(256), 0, stream>>>(v, mask, gated, rowmax, colmax,
                                                        rowsum, colsum, out);
}